// DKVMN_Model_27702539059713
// MI455X (gfx1250) — compile-verified
//
#include <hip/hip_runtime.h>
#include <hip/hip_bf16.h>
#include <math.h>

typedef float v2f __attribute__((ext_vector_type(2)));
typedef float v8f __attribute__((ext_vector_type(8)));

#define NUMQ 10000
#define NUMA 10000
#define DD   128
#define MM   50
#define BB   64
#define NN   200
#define ROWS (BB * NN)   /* 12800 */
#define WSTR 64          /* padded stride for w */

/* workspace layout (float offsets) */
#define K_OFF  0
#define S_OFF  (K_OFF + ROWS * DD)
#define E_OFF  (S_OFF + ROWS * DD)
#define A_OFF  (E_OFF + ROWS * DD)
#define W_OFF  (A_OFF + ROWS * DD)
#define R_OFF  (W_OFF + ROWS * WSTR)

__device__ __forceinline__ float sigmoidf_(float x) {
    return 1.0f / (1.0f + __expf(-x));
}

/* ---------------- Kernel 1: embedding gathers ----------------
   k = Kemb[q]; s = Vemb[q + NUMQ*r] + VAemb[a + NUMA*r]
   one wave per row, float4 vector loads. grid=1600, block=256.
   Embedding tables (25 MB total) stay L2-resident (RT policy). */
__global__ void gather_kernel(const int* __restrict__ q, const int* __restrict__ r,
                              const int* __restrict__ a,
                              const float* __restrict__ Kemb,
                              const float* __restrict__ Vemb,
                              const float* __restrict__ VAemb,
                              float* __restrict__ ws) {
    const int wave = threadIdx.x >> 5;
    const int lane = threadIdx.x & 31;
    const int row  = blockIdx.x * 8 + wave;
    const int qi = q[row], ri = r[row], ai = a[row];
    const int xi  = qi + NUMQ * ri;
    const int axi = ai + NUMA * ri;
    const float4 kv  = ((const float4*)Kemb )[qi  * 32 + lane];
    const float4 vv  = ((const float4*)Vemb )[xi  * 32 + lane];
    const float4 vav = ((const float4*)VAemb)[axi * 32 + lane];
    float4 sv;
    sv.x = vv.x + vav.x; sv.y = vv.y + vav.y;
    sv.z = vv.z + vav.z; sv.w = vv.w + vav.w;
    ((float4*)(ws + K_OFF))[row * 32 + lane] = kv;
    ((float4*)(ws + S_OFF))[row * 32 + lane] = sv;
}

/* ---------------- Kernel 2: erase/add gates ----------------
   e = sigmoid(s @ eW^T + eb), ad = tanh(s @ aW^T + ab)
   one wave per 16x16 output tile, both gates share the A operand.
   grid=800 (row tiles), block=256 (8 waves = 8 col tiles). */
__global__ void gates_kernel(const float* __restrict__ eW, const float* __restrict__ eb,
                             const float* __restrict__ aW, const float* __restrict__ ab,
                             float* __restrict__ ws) {
    const float* S = ws + S_OFF;
    float* E = ws + E_OFF;
    float* A = ws + A_OFF;
    const int lane = threadIdx.x & 31;
    const int wave = threadIdx.x >> 5;
    const int l15  = lane & 15;
    const int half = lane >> 4;
    const int r0 = blockIdx.x * 16;
    const int c0 = wave * 16;
    const float* srow = S  + (r0 + l15) * DD + 2 * half;
    const float* erow = eW + (c0 + l15) * DD + 2 * half;
    const float* arow = aW + (c0 + l15) * DD + 2 * half;
    v8f acc_e = {};
    v8f acc_a = {};
#pragma unroll
    for (int kk = 0; kk < DD; kk += 4) {
        v2f av = *(const v2f*)(srow + kk);
        v2f be = *(const v2f*)(erow + kk);
        v2f ba = *(const v2f*)(arow + kk);
        acc_e = __builtin_amdgcn_wmma_f32_16x16x4_f32(false, av, false, be,
                                                      (short)0, acc_e, false, false);
        acc_a = __builtin_amdgcn_wmma_f32_16x16x4_f32(false, av, false, ba,
                                                      (short)0, acc_a, false, false);
    }
    const float ebv = eb[c0 + l15];
    const float abv = ab[c0 + l15];
#pragma unroll
    for (int i = 0; i < 8; ++i) {
        const int gr = r0 + i + 8 * half;
        E[gr * DD + c0 + l15] = sigmoidf_(acc_e[i] + ebv);
        A[gr * DD + c0 + l15] = tanhf(acc_a[i] + abv);
    }
}

/* ---------------- Kernel 3: attention weights ----------------
   w = softmax(k @ Mk^T) over m=50 (columns padded to 64 with zeros
   via cndmask selects — no EXEC divergence around WMMA).
   grid=800, block=128 (4 waves = 4 col tiles of 16). */
__global__ void wsoft_kernel(const float* __restrict__ Mk, float* __restrict__ ws) {
    __shared__ float lds[16 * 65];
    const float* K = ws + K_OFF;
    float* W = ws + W_OFF;
    const int lane = threadIdx.x & 31;
    const int wave = threadIdx.x >> 5;
    const int l15  = lane & 15;
    const int half = lane >> 4;
    const int r0 = blockIdx.x * 16;
    const int c0 = wave * 16;
    const int col = c0 + l15;
    const int ccl = (col < MM) ? col : 0;        /* safe clamped address */
    const float* krow = K  + (r0 + l15) * DD + 2 * half;
    const float* mrow = Mk + ccl * DD + 2 * half;
    v8f acc = {};
#pragma unroll
    for (int kk = 0; kk < DD; kk += 4) {
        v2f av = *(const v2f*)(krow + kk);
        v2f bm = *(const v2f*)(mrow + kk);
        v2f bv;
        bv.x = (col < MM) ? bm.x : 0.0f;
        bv.y = (col < MM) ? bm.y : 0.0f;
        acc = __builtin_amdgcn_wmma_f32_16x16x4_f32(false, av, false, bv,
                                                    (short)0, acc, false, false);
    }
#pragma unroll
    for (int i = 0; i < 8; ++i)
        lds[(i + 8 * half) * 65 + col] = acc[i];
    __syncthreads();
    if (threadIdx.x < 16) {
        const int row = threadIdx.x;
        float mx = -3.4e38f;
        for (int j = 0; j < MM; ++j) mx = fmaxf(mx, lds[row * 65 + j]);
        float ssum = 0.0f;
        for (int j = 0; j < MM; ++j) {
            const float v = __expf(lds[row * 65 + j] - mx);
            lds[row * 65 + j] = v;
            ssum += v;
        }
        const float inv = 1.0f / ssum;
        for (int j = 0; j < MM; ++j)
            W[(r0 + row) * WSTR + j] = lds[row * 65 + j] * inv;
    }
}

/* ---------------- Kernel 4: memory scan ----------------
   Each block = (batch b, dim-quarter). One wave; each thread owns one
   dim and keeps all 50 memory cells in VGPRs. read = sum_m w*Mv is a
   thread-local 50-FMA reduction.
   The 327 MB Mv trajectory is written with NON-TEMPORAL stores
   (TH_STORE_NT) so the write-once stream does not evict the 192 MB L2;
   e/ad are read exactly once per element -> non-temporal loads.
   The w row is reused by the 4 dim-quarter blocks -> regular loads.
   grid=256, block=32. */
__global__ void scan_kernel(const float* __restrict__ Mv0,
                            float* __restrict__ ws, float* __restrict__ out) {
    const int b       = blockIdx.x >> 2;
    const int quarter = blockIdx.x & 3;
    const int lane    = threadIdx.x;
    const int dim     = quarter * 32 + lane;
    const float* Ew = ws + E_OFF;
    const float* Aw = ws + A_OFF;
    const float* Ww = ws + W_OFF;
    float* Rw = ws + R_OFF;
    float* MvOut = out + ROWS + (size_t)b * (NN + 1) * MM * DD;

    float Mreg[MM];
#pragma unroll
    for (int m = 0; m < MM; ++m) {
        Mreg[m] = Mv0[m * DD + dim];
        __builtin_nontemporal_store(Mreg[m], MvOut + m * DD + dim);  /* t=0 slice */
    }
    __shared__ float wl[64];
    for (int t = 0; t < NN; ++t) {
        const int row = b * NN + t;
        wl[lane] = Ww[row * WSTR + lane];
        if (lane < MM - 32) wl[32 + lane] = Ww[row * WSTR + 32 + lane];
        __syncthreads();
        const float et = __builtin_nontemporal_load(Ew + row * DD + dim);
        const float at = __builtin_nontemporal_load(Aw + row * DD + dim);
        float rd = 0.0f;
#pragma unroll
        for (int m = 0; m < MM; ++m) rd += wl[m] * Mreg[m];   /* pre-update read */
        Rw[row * DD + dim] = rd;
        float* dst = MvOut + (size_t)(t + 1) * MM * DD;
#pragma unroll
        for (int m = 0; m < MM; ++m) {
            const float wm = wl[m];
            Mreg[m] = Mreg[m] * (1.0f - wm * et) + wm * at;
            __builtin_nontemporal_store(Mreg[m], dst + m * DD + dim);
        }
        __syncthreads();
    }
}

/* ---------------- Kernel 5: output head ----------------
   f = tanh([reads,k] @ fW^T + fb)   (K = 256, A switches source at 128)
   p = sigmoid(f @ pW^T + pb)
   grid=800, block=256 (8 waves = 8 col tiles). */
__global__ void final_kernel(const float* __restrict__ fW, const float* __restrict__ fb,
                             const float* __restrict__ pW, const float* __restrict__ pb,
                             float* __restrict__ ws, float* __restrict__ out) {
    __shared__ float fs[16 * 129];
    const float* Rd = ws + R_OFF;
    const float* K  = ws + K_OFF;
    const int lane = threadIdx.x & 31;
    const int wave = threadIdx.x >> 5;
    const int l15  = lane & 15;
    const int half = lane >> 4;
    const int r0 = blockIdx.x * 16;
    const int c0 = wave * 16;
    const float* rrow = Rd + (r0 + l15) * DD + 2 * half;
    const float* krow = K  + (r0 + l15) * DD + 2 * half;
    const float* wrow = fW + (c0 + l15) * (2 * DD) + 2 * half;
    v8f acc = {};
#pragma unroll
    for (int kk = 0; kk < 2 * DD; kk += 4) {
        v2f av = (kk < DD) ? *(const v2f*)(rrow + kk)
                           : *(const v2f*)(krow + kk - DD);
        v2f bv = *(const v2f*)(wrow + kk);
        acc = __builtin_amdgcn_wmma_f32_16x16x4_f32(false, av, false, bv,
                                                    (short)0, acc, false, false);
    }
    const float bias = fb[c0 + l15];
#pragma unroll
    for (int i = 0; i < 8; ++i)
        fs[(i + 8 * half) * 129 + c0 + l15] = tanhf(acc[i] + bias);
    __syncthreads();
    if (threadIdx.x < 16) {
        const int row = threadIdx.x;
        float s = pb[0];
        for (int j = 0; j < DD; ++j) s += fs[row * 129 + j] * pW[j];
        out[r0 + row] = sigmoidf_(s);
    }
}

extern "C" void kernel_launch(void* const* d_in, const int* in_sizes, int n_in,
                              void* d_out, int out_size, void* d_ws, size_t ws_size,
                              hipStream_t stream) {
    (void)in_sizes; (void)n_in; (void)out_size; (void)ws_size;
    const int*   q     = (const int*)  d_in[0];
    const int*   r     = (const int*)  d_in[1];
    const int*   a     = (const int*)  d_in[2];
    const float* Kemb  = (const float*)d_in[3];
    const float* Vemb  = (const float*)d_in[4];
    const float* VAemb = (const float*)d_in[5];
    const float* Mk    = (const float*)d_in[6];
    const float* Mv0   = (const float*)d_in[7];
    const float* fW    = (const float*)d_in[8];
    const float* fb    = (const float*)d_in[9];
    const float* pW    = (const float*)d_in[10];
    const float* pb    = (const float*)d_in[11];
    const float* eW    = (const float*)d_in[12];
    const float* eb    = (const float*)d_in[13];
    const float* aW    = (const float*)d_in[14];
    const float* ab    = (const float*)d_in[15];
    float* out = (float*)d_out;
    float* ws  = (float*)d_ws;

    gather_kernel<<<ROWS / 8, 256, 0, stream>>>(q, r, a, Kemb, Vemb, VAemb, ws);
    gates_kernel <<<ROWS / 16, 256, 0, stream>>>(eW, eb, aW, ab, ws);
    wsoft_kernel <<<ROWS / 16, 128, 0, stream>>>(Mk, ws);
    scan_kernel  <<<BB * 4, 32, 0, stream>>>(Mv0, ws, out);
    final_kernel <<<ROWS / 16, 256, 0, stream>>>(fW, fb, pW, pb, ws, out);
}